// MultiHeadDotProduct_4277787427366
// MI455X (gfx1250) — compile-verified
//
#include <hip/hip_runtime.h>
#include <hip/hip_bf16.h>

// Graph multi-head attention for MI455X (gfx1250), wave32.
// q/k/v/accum are L2-resident (192MB L2 >> 102MB working set) so the
// random per-edge gathers and the f32-atomic scatter run at L2 speed.
// GEMMs use exact-precision V_WMMA_F32_16X16X4_F32.

typedef float v2f __attribute__((ext_vector_type(2)));
typedef float v8f __attribute__((ext_vector_type(8)));

#define EMBED 128
#define HEADS 4
#define HEAD_DIM 32

// ---------------------------------------------------------------------------
// C[nrows,128] = A[nrows,128] @ W^T + bias   (W row-major [128,128])
// One block = 16 rows of A staged in LDS; 8 waves, one 16x16 tile each.
// Requires nrows % 16 == 0 (true: 50000 = 3125*16).
// ---------------------------------------------------------------------------
__global__ void gemm128_wmma(const float* __restrict__ A,
                             const float* __restrict__ W,
                             const float* __restrict__ bias,
                             float* __restrict__ C) {
  __shared__ float sA[16 * EMBED];  // 8 KB
  const int row0 = blockIdx.x * 16;
  const int tid = threadIdx.x;  // 256 threads = 8 waves

  // Stage 16x128 fp32 A tile: 512 float4, 2 per thread.
  {
    const float4* src = (const float4*)(A + (size_t)row0 * EMBED);
    float4* dst = (float4*)sA;
    dst[tid] = src[tid];
    dst[tid + 256] = src[tid + 256];
  }
  __syncthreads();

  const int wave = tid >> 5;
  const int lane = tid & 31;
  const int n0 = wave * 16;          // output column tile
  const int m = lane & 15;           // row within tile (A) / col within tile (B,C)
  const int koff = (lane >> 4) << 1; // K sub-offset: lanes 0-15 -> K+0/1, 16-31 -> K+2/3

  // A frag: lane holds A[m][kk+koff], A[m][kk+koff+1]
  const float* ap = sA + m * EMBED + koff;
  // B frag: B[k][n] = W[n][k]; lane holds W[n0+m][kk+koff], W[n0+m][kk+koff+1]
  const int wrow = n0 + m;
  const float* wp = W + (size_t)wrow * EMBED + koff;

  v8f acc = {};
#pragma unroll
  for (int kk = 0; kk < EMBED; kk += 4) {
    v2f a = *(const v2f*)(ap + kk);
    v2f b = *(const v2f*)(wp + kk);
    acc = __builtin_amdgcn_wmma_f32_16x16x4_f32(
        /*neg_a=*/false, a, /*neg_b=*/false, b,
        /*c_mod=*/(short)0, acc, /*reuse_a=*/false, /*reuse_b=*/false);
  }

  const float bn = bias[wrow];
  const int moff = (lane >> 4) << 3;  // lanes 16-31 hold M = r+8
  float* out = C + (size_t)(row0 + moff) * EMBED + wrow;
#pragma unroll
  for (int r = 0; r < 8; ++r) out[(size_t)r * EMBED] = acc[r] + bn;
}

// ---------------------------------------------------------------------------
// Zero the attention accumulator; segmax = -inf, denom = 0.
// ---------------------------------------------------------------------------
__global__ void init_buffers(float* __restrict__ accum,
                             float* __restrict__ segmax,
                             float* __restrict__ denom, int n) {
  const int idx = blockIdx.x * blockDim.x + threadIdx.x;
  if (idx < n * EMBED) accum[idx] = 0.0f;
  if (idx < n * HEADS) {
    segmax[idx] = -__builtin_inff();
    denom[idx] = 0.0f;
  }
}

// ---------------------------------------------------------------------------
// scores[e,h] = dot(q[col[e],h,:], k[row[e],h,:]) / sqrt(32); atomicMax by row.
// One wave32 per edge: lane loads float4 (128 ch / 32 lanes); 8-lane head
// groups reduced with shfl_xor (wave32 correct).
// ---------------------------------------------------------------------------
__global__ void edge_scores(const float* __restrict__ q,
                            const float* __restrict__ k,
                            const long long* __restrict__ ei,
                            float* __restrict__ scores,
                            float* __restrict__ segmax, int E) {
  const int e = (blockIdx.x * blockDim.x + threadIdx.x) >> 5;
  if (e >= E) return;
  const int lane = threadIdx.x & 31;
  const long long row = ei[2 * (long long)e];
  const long long col = ei[2 * (long long)e + 1];

  const float4 qv = ((const float4*)(q + col * EMBED))[lane];
  const float4 kv = ((const float4*)(k + row * EMBED))[lane];
  float p = qv.x * kv.x + qv.y * kv.y + qv.z * kv.z + qv.w * kv.w;
  // reduce within each 8-lane head group
  p += __shfl_xor(p, 1, 32);
  p += __shfl_xor(p, 2, 32);
  p += __shfl_xor(p, 4, 32);
  if ((lane & 7) == 0) {
    const int h = lane >> 3;
    const float s = p * 0.17677669529663687f;  // 1/sqrt(HEAD_DIM)
    scores[(size_t)e * HEADS + h] = s;
    atomicMax(&segmax[row * HEADS + h], s);
  }
}

// ---------------------------------------------------------------------------
// ex = exp(score - segmax[row,h]); denom[row,h] += ex. (in-place on scores)
// ---------------------------------------------------------------------------
__global__ void edge_exp(const long long* __restrict__ ei,
                         float* __restrict__ scores,
                         const float* __restrict__ segmax,
                         float* __restrict__ denom, int E) {
  const int idx = blockIdx.x * blockDim.x + threadIdx.x;
  if (idx >= E * HEADS) return;
  const int e = idx >> 2;
  const int h = idx & 3;
  const long long row = ei[2 * (long long)e];
  float m = segmax[row * HEADS + h];
  if (m == -__builtin_inff()) m = 0.0f;  // reference: non-finite -> 0
  const float x = __expf(scores[idx] - m);
  scores[idx] = x;
  atomicAdd(&denom[row * HEADS + h], x);
}

// ---------------------------------------------------------------------------
// accum[col] += (ex/denom[row,h]) * v[row].  Wave per edge; 4 atomics/lane
// to consecutive addresses -> coalesced L2 atomics.
// ---------------------------------------------------------------------------
__global__ void edge_scatter(const float* __restrict__ vfeat,
                             const long long* __restrict__ ei,
                             const float* __restrict__ ex,
                             const float* __restrict__ denom,
                             float* __restrict__ accum, int E) {
  const int e = (blockIdx.x * blockDim.x + threadIdx.x) >> 5;
  if (e >= E) return;
  const int lane = threadIdx.x & 31;
  const long long row = ei[2 * (long long)e];
  const long long col = ei[2 * (long long)e + 1];
  const int h = lane >> 3;
  const float a = ex[(size_t)e * HEADS + h] / denom[row * HEADS + h];
  const float4 vv = ((const float4*)(vfeat + row * EMBED))[lane];
  float* dst = accum + col * EMBED + lane * 4;
  atomicAdd(dst + 0, a * vv.x);
  atomicAdd(dst + 1, a * vv.y);
  atomicAdd(dst + 2, a * vv.z);
  atomicAdd(dst + 3, a * vv.w);
}

// ---------------------------------------------------------------------------
extern "C" void kernel_launch(void* const* d_in, const int* in_sizes, int n_in,
                              void* d_out, int out_size, void* d_ws, size_t ws_size,
                              hipStream_t stream) {
  const float* feats = (const float*)d_in[0];
  const long long* edge_index = (const long long*)d_in[1];
  const float* edge_attr = (const float*)d_in[2];
  const float* Wq = (const float*)d_in[3];
  const float* bq = (const float*)d_in[4];
  const float* Wk = (const float*)d_in[5];
  const float* bk = (const float*)d_in[6];
  const float* Wv = (const float*)d_in[7];
  const float* bv = (const float*)d_in[8];
  const float* Wo = (const float*)d_in[9];
  const float* bo = (const float*)d_in[10];

  const int N = in_sizes[0] / EMBED;  // 50000
  const int E = in_sizes[1] / 2;      // 800000

  // Workspace layout (floats): q | k | v | accum | scores(ex) | segmax | denom
  float* q = (float*)d_ws;
  float* k = q + (size_t)N * EMBED;
  float* v = k + (size_t)N * EMBED;
  float* accum = v + (size_t)N * EMBED;
  float* scores = accum + (size_t)N * EMBED;
  float* segmax = scores + (size_t)E * HEADS;
  float* denom = segmax + (size_t)N * HEADS;
  // total ~117 MB

  const int gemm_blocks = N / 16;  // N % 16 == 0

  init_buffers<<<(N * EMBED + 255) / 256, 256, 0, stream>>>(accum, segmax, denom, N);

  gemm128_wmma<<<gemm_blocks, 256, 0, stream>>>(feats, Wq, bq, q);
  gemm128_wmma<<<gemm_blocks, 256, 0, stream>>>(feats, Wk, bk, k);
  gemm128_wmma<<<gemm_blocks, 256, 0, stream>>>(feats, Wv, bv, v);

  edge_scores<<<(E + 7) / 8, 256, 0, stream>>>(q, k, edge_index, scores, segmax, E);
  edge_exp<<<(E * HEADS + 255) / 256, 256, 0, stream>>>(edge_index, scores, segmax, denom, E);
  edge_scatter<<<(E + 7) / 8, 256, 0, stream>>>(v, edge_index, scores, denom, accum, E);

  gemm128_wmma<<<gemm_blocks, 256, 0, stream>>>(accum, Wo, bo, (float*)d_out);

  // Pass-through tuple outputs: edge_index (int64) then edge_attr (f32),
  // byte-concatenated after feats_out.
  char* outb = (char*)d_out;
  size_t off = (size_t)N * EMBED * sizeof(float);
  hipMemcpyAsync(outb + off, edge_index, (size_t)E * 2 * sizeof(long long),
                 hipMemcpyDeviceToDevice, stream);
  off += (size_t)E * 2 * sizeof(long long);
  hipMemcpyAsync(outb + off, edge_attr, (size_t)E * sizeof(float),
                 hipMemcpyDeviceToDevice, stream);
}